// SimpleMinkUNet_9380208575023
// MI455X (gfx1250) — compile-verified
//
#include <hip/hip_runtime.h>

// -------- WMMA fragment types (wave32, gfx1250) --------
typedef __attribute__((ext_vector_type(2))) float v2f;   // A/B frag of V_WMMA_F32_16X16X4_F32
typedef __attribute__((ext_vector_type(8))) float v8f;   // 16x16 f32 C/D frag
typedef __attribute__((ext_vector_type(4))) float v4f;

// ---------------------------------------------------------------------------
// vectorized zero fill (count must be a multiple of 4)
// ---------------------------------------------------------------------------
__global__ void zero_f32v4(float* __restrict__ p, long long count4) {
  long long i = (long long)blockIdx.x * blockDim.x + threadIdx.x;
  if (i < count4) {
    v4f z = {};
    *(v4f*)(p + 4 * i) = z;                 // global_store_b128
  }
}

// ---------------------------------------------------------------------------
// Sparse conv: persistent one-wave blocks. Each block owns (kernel offset k,
// 16-wide output-channel tile) and sweeps TPB pair-tiles, keeping the B
// fragments of W[k] register-resident. Per tile: uniform early-skip of
// all-sentinel padding tiles (ballot), gather A rows, chained f32 WMMA
// (K=4) over Cin, scatter-add rows via global_atomic_add_f32.
//
// grid = (ceil(P/(16*TPB)), 27, ceil(Cout/16)), block = 32 (one wave)
//
// A layout (32-bit A 16x4, ISA 7.12.2): lanes 0-15 -> M=lane, V0=K0,V1=K1;
//                                       lanes 16-31 -> M=lane-16, V0=K2,V1=K3.
// B layout (4x16):                      lanes 0-15 -> N=lane, K0/K1;
//                                       lanes 16-31 -> N=lane-16, K2/K3.
// C/D layout: VGPR r -> lanes 0-15: (M=r, N=lane); lanes 16-31: (M=8+r, N=lane-16).
// ---------------------------------------------------------------------------
template <int CIN, int COUT, int TPB>
__global__ __launch_bounds__(32) void sconv_wmma(
    const float* __restrict__ f, const float* __restrict__ W,
    const int* __restrict__ ii, const int* __restrict__ oo,
    const int* __restrict__ nin_p, const int* __restrict__ nout_p,
    int P, float* __restrict__ out) {
  const int lane = threadIdx.x;
  const int m    = lane & 15;   // M for A, N for B/C/D
  const int half = lane >> 4;

  const int k       = blockIdx.y;          // kernel offset 0..26
  const int colBase = blockIdx.z << 4;

  const int n_in  = *nin_p;                // device scalars (n1/n2 unknown on host)
  const int n_out = *nout_p;

  const int col  = colBase + m;
  const bool cv  = (col < COUT);           // Cout=20 tail guard

  // ---- register-resident B fragments of W[k] for this 16-col tile ----
  const float* __restrict__ Wk = W + (long long)k * (CIN * COUT);
  v2f bfrag[CIN / 4];
#pragma unroll
  for (int c = 0; c < CIN / 4; ++c) {
    const int k0 = c * 4 + half * 2;
    if (cv) {
      bfrag[c].x = Wk[(k0 + 0) * COUT + col];
      bfrag[c].y = Wk[(k0 + 1) * COUT + col];
    } else {
      bfrag[c].x = 0.0f; bfrag[c].y = 0.0f;
    }
  }

  const int* __restrict__ iik = ii + (long long)k * P;
  const int* __restrict__ ook = oo + (long long)k * P;

#pragma unroll 1
  for (int t = 0; t < TPB; ++t) {
    const int tile = blockIdx.x * TPB + t;
    const int base = tile * 16;
    if (base >= P) break;
    const int p = base + m;
    const bool pv = (p < P);

    // prefetch next tile's index words (global_prefetch_b8)
    if (base + 16 < P) {
      __builtin_prefetch(iik + p + 16, 0, 1);
      __builtin_prefetch(ook + p + 16, 0, 1);
    }

    // output row for pair M=m; sentinel (>= n_out) rows are dropped
    const int o   = pv ? ook[p] : n_out;
    const bool ov = (o >= 0) && (o < n_out);
    if (__ballot(ov) == 0ull) continue;    // all-padding tile: uniform skip

    int row       = pv ? iik[p] : n_in;    // sentinel row gathers zeros
    const bool rv = (row >= 0) && (row < n_in);
    const float* __restrict__ fr = f + (long long)row * CIN;

    v8f acc = {};
#pragma unroll
    for (int c = 0; c < CIN / 4; ++c) {
      const int k0 = c * 4 + half * 2;
      v2f a;
      if (rv) {
        a = *(const v2f*)(fr + k0);        // 8B-aligned global_load_b64
      } else {
        a.x = 0.0f; a.y = 0.0f;
      }
      acc = __builtin_amdgcn_wmma_f32_16x16x4_f32(
          /*neg_a=*/false, a, /*neg_b=*/false, bfrag[c],
          /*c_mod=*/(short)0, acc, /*reuse_a=*/false, /*reuse_b=*/false);
    }

    // Scatter: lane covers rows M = half*8 + r, column N = m.
    // Row mr's output index is held by lane mr -> fetch via ds_bpermute.
#pragma unroll
    for (int r = 0; r < 8; ++r) {
      const int mr  = half * 8 + r;
      const int o_r = __shfl(o, mr);
      if (cv && base + mr < P && o_r >= 0 && o_r < n_out) {
        unsafeAtomicAdd(&out[(long long)o_r * COUT + col], acc[r]);
      }
    }
  }
}

// ---------------------------------------------------------------------------
// BN stats: thread (rgrp, c) accumulates a grid-stride slice of rows for
// channel c in registers, LDS tree-reduce across row-groups, one global
// atomic per channel per block. stats = [sum[0..C-1], sumsq[0..C-1]].
// C is a power of two (16/32/64).
// ---------------------------------------------------------------------------
template <int C>
__global__ __launch_bounds__(256) void bn_stats(
    const float* __restrict__ x, const int* __restrict__ n_p,
    float* __restrict__ stats) {
  constexpr int RPB = 256 / C;             // row-groups per block
  const int n = *n_p;
  const int t = threadIdx.x;
  const int c = t & (C - 1);
  const int rg = t / C;

  float s = 0.0f, q = 0.0f;
  for (long long row = (long long)blockIdx.x * RPB + rg; row < n;
       row += (long long)gridDim.x * RPB) {
    const float v = x[row * C + c];
    s += v;
    q += v * v;
  }

  __shared__ float ls[256];
  __shared__ float lq[256];
  ls[t] = s; lq[t] = q;
  __syncthreads();
#pragma unroll
  for (int off = 128; off >= C; off >>= 1) {
    if (t < off) { ls[t] += ls[t + off]; lq[t] += lq[t + off]; }
    __syncthreads();
  }
  if (t < C) {
    unsafeAtomicAdd(&stats[c], ls[t]);
    unsafeAtomicAdd(&stats[C + c], lq[t]);
  }
}

// ---------------------------------------------------------------------------
// BN normalize (training-mode, population variance) + ReLU, in place.
// ---------------------------------------------------------------------------
template <int C>
__global__ __launch_bounds__(256) void bn_relu(
    float* __restrict__ x, const float* __restrict__ stats,
    const float* __restrict__ g, const float* __restrict__ b,
    const int* __restrict__ n_p) {
  const int n = *n_p;
  const long long idx = (long long)blockIdx.x * 256 + threadIdx.x;
  const long long row = idx / C;
  const int c         = (int)(idx % C);
  if (row >= n) return;
  const float inv_n = 1.0f / (float)n;
  const float mean  = stats[c] * inv_n;
  const float var   = stats[C + c] * inv_n - mean * mean;
  const float y = (x[idx] - mean) * rsqrtf(var + 1e-5f) * g[c] + b[c];
  x[idx] = y > 0.0f ? y : 0.0f;
}

// ---------------------------------------------------------------------------
// Host orchestration
// ---------------------------------------------------------------------------
extern "C" void kernel_launch(void* const* d_in, const int* in_sizes, int n_in,
                              void* d_out, int out_size, void* d_ws, size_t ws_size,
                              hipStream_t stream) {
  (void)n_in; (void)ws_size;
  const float* feats = (const float*)d_in[0];
  const float* W0  = (const float*)d_in[1];
  const float* g0  = (const float*)d_in[2];
  const float* b0  = (const float*)d_in[3];
  const float* W1  = (const float*)d_in[4];
  const float* g1  = (const float*)d_in[5];
  const float* b1  = (const float*)d_in[6];
  const float* W2  = (const float*)d_in[7];
  const float* g2  = (const float*)d_in[8];
  const float* b2  = (const float*)d_in[9];
  const float* Wt1 = (const float*)d_in[10];
  const float* gt1 = (const float*)d_in[11];
  const float* bt1 = (const float*)d_in[12];
  const float* Wt2 = (const float*)d_in[13];
  const int* in0  = (const int*)d_in[14];
  const int* out0 = (const int*)d_in[15];
  const int* in1  = (const int*)d_in[16];
  const int* out1 = (const int*)d_in[17];
  const int* in2  = (const int*)d_in[18];
  const int* out2 = (const int*)d_in[19];
  const int* int1 = (const int*)d_in[20];
  const int* outt1= (const int*)d_in[21];
  const int* int2 = (const int*)d_in[22];
  const int* outt2= (const int*)d_in[23];
  const int* n0p  = (const int*)d_in[24];
  const int* n1p  = (const int*)d_in[25];
  const int* n2p  = (const int*)d_in[26];

  const int n0  = in_sizes[0] / 4;     // host-known; n1,n2 only as device scalars
  const int P0  = in_sizes[14] / 27;
  const int P1  = in_sizes[16] / 27;
  const int P2  = in_sizes[18] / 27;
  const int Pt1 = in_sizes[20] / 27;
  const int Pt2 = in_sizes[22] / 27;

  // workspace layout (floats); n1,n2 <= n0 so n0-row bounds are safe
  float* ws  = (float*)d_ws;
  float* x0  = ws;                          // n0 * 16
  float* x1  = x0  + (size_t)n0 * 16;       // <=n0 * 32
  float* x2  = x1  + (size_t)n0 * 32;       // <=n0 * 64
  float* xt1 = x2  + (size_t)n0 * 64;       // <=n0 * 32
  float* st0 = xt1 + (size_t)n0 * 32;       // 32
  float* st1 = st0 + 32;                    // 64
  float* st2 = st1 + 64;                    // 128
  float* stt = st2 + 128;                   // 64

  // zero accumulators + stats (n0*144 is a multiple of 4) and final output
  const long long zc4 = ((long long)n0 * (16 + 32 + 64 + 32) + 512) / 4;
  zero_f32v4<<<dim3((unsigned)((zc4 + 255) / 256)), 256, 0, stream>>>(ws, zc4);
  const long long oc4 = ((long long)out_size + 3) / 4;   // n0*20, multiple of 4
  zero_f32v4<<<dim3((unsigned)((oc4 + 255) / 256)), 256, 0, stream>>>(
      (float*)d_out, oc4);

  constexpr int TPB = 8;                   // pair-tiles per persistent block
  const dim3 wv(32);
  auto tgrid = [](int P) { return (unsigned)((P + 16 * TPB - 1) / (16 * TPB)); };
  auto eg = [](long long elems) { return dim3((unsigned)((elems + 255) / 256)); };
  auto sg = [&](int C) {                   // bn_stats grid (bounded, deterministic)
    long long blocks = ((long long)n0 * C + 255) / 256;
    return dim3((unsigned)(blocks < 768 ? blocks : 768));
  };

  // conv0: 4 -> 16 (stride 1, n0 -> n0)
  sconv_wmma<4, 16, TPB><<<dim3(tgrid(P0), 27, 1), wv, 0, stream>>>(
      feats, W0, in0, out0, n0p, n0p, P0, x0);
  bn_stats<16><<<sg(16), 256, 0, stream>>>(x0, n0p, st0);
  bn_relu<16><<<eg((long long)n0 * 16), 256, 0, stream>>>(x0, st0, g0, b0, n0p);

  // conv1: 16 -> 32 (n0 -> n1)
  sconv_wmma<16, 32, TPB><<<dim3(tgrid(P1), 27, 2), wv, 0, stream>>>(
      x0, W1, in1, out1, n0p, n1p, P1, x1);
  bn_stats<32><<<sg(32), 256, 0, stream>>>(x1, n1p, st1);
  bn_relu<32><<<eg((long long)n0 * 32), 256, 0, stream>>>(x1, st1, g1, b1, n1p);

  // conv2: 32 -> 64 (n1 -> n2)
  sconv_wmma<32, 64, TPB><<<dim3(tgrid(P2), 27, 4), wv, 0, stream>>>(
      x1, W2, in2, out2, n1p, n2p, P2, x2);
  bn_stats<64><<<sg(64), 256, 0, stream>>>(x2, n2p, st2);
  bn_relu<64><<<eg((long long)n0 * 64), 256, 0, stream>>>(x2, st2, g2, b2, n2p);

  // convtr1: 64 -> 32 (n2 -> n1)
  sconv_wmma<64, 32, TPB><<<dim3(tgrid(Pt1), 27, 2), wv, 0, stream>>>(
      x2, Wt1, int1, outt1, n2p, n1p, Pt1, xt1);
  bn_stats<32><<<sg(32), 256, 0, stream>>>(xt1, n1p, stt);
  bn_relu<32><<<eg((long long)n0 * 32), 256, 0, stream>>>(xt1, stt, gt1, bt1, n1p);

  // convtr2: 32 -> 20 (n1 -> n0), straight into d_out (col tile 2 is masked)
  sconv_wmma<32, 20, TPB><<<dim3(tgrid(Pt2), 27, 2), wv, 0, stream>>>(
      xt1, Wt2, int2, outt2, n1p, n0p, Pt2, (float*)d_out);
}